// GINEncoder_72859825209483
// MI455X (gfx1250) — compile-verified
//
#include <hip/hip_runtime.h>

typedef float v2f __attribute__((ext_vector_type(2)));
typedef float v8f __attribute__((ext_vector_type(8)));

#define IN_FEATS 128
#define HIDDEN   64
#define NGRAPH   512
#define BN_EPS   1e-5f

// -------------------- edge scatter-add aggregation --------------------
// agg must be pre-initialized with x (gives h = x + sum_neighbors).
// One thread per (edge, 4-feature chunk). float4 gather, 4x f32 atomics.
__global__ __launch_bounds__(256) void scatter_add_kernel(
    const float* __restrict__ xin, const int* __restrict__ ei,
    float* __restrict__ agg, int E, int F) {
  long long tid = (long long)blockIdx.x * blockDim.x + threadIdx.x;
  int chunks = F >> 2;
  long long total = (long long)E * chunks;
  if (tid >= total) return;
  int e = (int)(tid / chunks);
  int f = ((int)(tid - (long long)e * chunks)) << 2;
  int s = ei[e];
  int d = ei[E + e];
  const float4 vv = *(const float4*)(xin + (long long)s * F + f);
  float* p = agg + (long long)d * F + f;
  atomicAdd(p + 0, vv.x);
  atomicAdd(p + 1, vv.y);
  atomicAdd(p + 2, vv.z);
  atomicAdd(p + 3, vv.w);
}

// -------------------- fused GIN MLP: relu(BN(relu(h@Wa^T+ba)@Wb^T+bb)) --------------------
// One wave (32 lanes) owns a 16-node row tile; 4 column tiles of 16 cover HIDDEN=64.
// fp32 WMMA 16x16x4. A frag: lanes 0-15 K={k,k+1}, lanes 16-31 K={k+2,k+3}.
template <int FIN>
__global__ __launch_bounds__(256) void gin_mlp_kernel(
    const float* __restrict__ hin,  // N x FIN
    const float* __restrict__ Wa,   // 64 x FIN (row major)
    const float* __restrict__ ba,   // 64
    const float* __restrict__ Wb,   // 64 x 64
    const float* __restrict__ bb,   // 64
    const float* __restrict__ gg,   // BN gamma
    const float* __restrict__ be,   // BN beta
    const float* __restrict__ mm,   // BN mean
    const float* __restrict__ vv,   // BN var
    float* __restrict__ hout,       // N x 64
    int n_nodes) {
  __shared__ float sT[8][16 * 64];  // per-wave 16x64 intermediate (re-layout buffer)

  const int wave = threadIdx.x >> 5;
  const int lane = threadIdx.x & 31;
  const int half = lane >> 4;  // 0 -> lanes 0-15, 1 -> lanes 16-31
  const int l16  = lane & 15;

  const int tile = blockIdx.x * 8 + wave;
  const int row0 = tile * 16;
  if (row0 >= n_nodes) return;  // wave-uniform: EXEC all-1s for surviving waves

  // ---- GEMM1: acc = hin_tile(16xFIN) @ Wa^T + ba ----
  v8f acc[4];
#pragma unroll
  for (int t = 0; t < 4; ++t) {
    float bias = ba[t * 16 + l16];  // C/D: column depends only on (lane&15)
#pragma unroll
    for (int r = 0; r < 8; ++r) acc[t][r] = bias;
  }

  int arow = row0 + l16;
  if (arow >= n_nodes) arow = n_nodes - 1;  // clamp: compute junk rows, never store them
  const float* aptr = hin + (long long)arow * FIN + 2 * half;

  for (int k = 0; k < FIN; k += 4) {
    v2f a = *(const v2f*)(aptr + k);
#pragma unroll
    for (int t = 0; t < 4; ++t) {
      const float* bp = Wa + (t * 16 + l16) * FIN + k + 2 * half;  // B[k][n] = Wa[n][k]
      v2f b = *(const v2f*)bp;
      acc[t] = __builtin_amdgcn_wmma_f32_16x16x4_f32(
          false, a, false, b, (short)0, acc[t], false, false);
    }
  }

  // ---- ReLU; spill 16x64 tile to LDS to convert C-layout -> A-layout ----
#pragma unroll
  for (int t = 0; t < 4; ++t) {
#pragma unroll
    for (int r = 0; r < 8; ++r) {
      float x = acc[t][r];
      x = x > 0.0f ? x : 0.0f;
      sT[wave][(r + 8 * half) * 64 + t * 16 + l16] = x;
    }
  }
  // same-wave DS RAW; compiler inserts s_wait_dscnt

  // ---- GEMM2: acc2 = T(16x64) @ Wb^T ----
  v8f acc2[4];
#pragma unroll
  for (int t = 0; t < 4; ++t)
#pragma unroll
    for (int r = 0; r < 8; ++r) acc2[t][r] = 0.0f;

  const float* tp = &sT[wave][l16 * 64 + 2 * half];
  for (int k = 0; k < 64; k += 4) {
    v2f a = *(const v2f*)(tp + k);
#pragma unroll
    for (int t = 0; t < 4; ++t) {
      const float* bp = Wb + (t * 16 + l16) * 64 + k + 2 * half;
      v2f b = *(const v2f*)bp;
      acc2[t] = __builtin_amdgcn_wmma_f32_16x16x4_f32(
          false, a, false, b, (short)0, acc2[t], false, false);
    }
  }

  // ---- fused bias + BN(eval) + ReLU:  out = relu(raw*s + ((bb-m)*s + be)) ----
#pragma unroll
  for (int t = 0; t < 4; ++t) {
    int n = t * 16 + l16;
    float s  = gg[n] * rsqrtf(vv[n] + BN_EPS);
    float sh = (bb[n] - mm[n]) * s + be[n];
#pragma unroll
    for (int r = 0; r < 8; ++r) {
      int mrow = row0 + r + 8 * half;
      float o = acc2[t][r] * s + sh;
      o = o > 0.0f ? o : 0.0f;
      if (mrow < n_nodes) hout[(long long)mrow * 64 + n] = o;
    }
  }
}

// -------------------- global mean pool --------------------
__global__ __launch_bounds__(256) void pool_zero_kernel(float* __restrict__ sums,
                                                        float* __restrict__ counts) {
  int i = blockIdx.x * blockDim.x + threadIdx.x;
  if (i < NGRAPH * HIDDEN) sums[i] = 0.0f;
  if (i < NGRAPH) counts[i] = 0.0f;
}

__global__ __launch_bounds__(256) void pool_sum_kernel(
    const float* __restrict__ h, const int* __restrict__ batch,
    float* __restrict__ sums, float* __restrict__ counts, int n_nodes) {
  long long tid = (long long)blockIdx.x * blockDim.x + threadIdx.x;
  long long total = (long long)n_nodes * 16;
  if (tid >= total) return;
  int i = (int)(tid >> 4);
  int c = ((int)tid & 15) << 2;
  int g = batch[i];
  const float4 vv = *(const float4*)(h + (long long)i * HIDDEN + c);
  float* p = sums + (long long)g * HIDDEN + c;
  atomicAdd(p + 0, vv.x);
  atomicAdd(p + 1, vv.y);
  atomicAdd(p + 2, vv.z);
  atomicAdd(p + 3, vv.w);
  if (c == 0) atomicAdd(&counts[g], 1.0f);
}

__global__ __launch_bounds__(256) void pool_div_kernel(
    const float* __restrict__ sums, const float* __restrict__ counts,
    float* __restrict__ out) {
  int i = blockIdx.x * blockDim.x + threadIdx.x;
  if (i >= NGRAPH * HIDDEN) return;
  float c = counts[i >> 6];
  c = c > 1.0f ? c : 1.0f;
  out[i] = sums[i] / c;
}

// -------------------- host launch --------------------
extern "C" void kernel_launch(void* const* d_in, const int* in_sizes, int n_in,
                              void* d_out, int out_size, void* d_ws, size_t ws_size,
                              hipStream_t stream) {
  const float* x   = (const float*)d_in[0];
  const int* ei    = (const int*)d_in[1];    // (2, E) flat: [src | dst]
  const int* batch = (const int*)d_in[2];
  const float* W1a = (const float*)d_in[3];
  const float* b1a = (const float*)d_in[4];
  const float* W1b = (const float*)d_in[5];
  const float* b1b = (const float*)d_in[6];
  const float* g1  = (const float*)d_in[7];
  const float* be1 = (const float*)d_in[8];
  const float* m1  = (const float*)d_in[9];
  const float* v1  = (const float*)d_in[10];
  const float* W2a = (const float*)d_in[11];
  const float* b2a = (const float*)d_in[12];
  const float* W2b = (const float*)d_in[13];
  const float* b2b = (const float*)d_in[14];
  const float* g2  = (const float*)d_in[15];
  const float* be2 = (const float*)d_in[16];
  const float* m2  = (const float*)d_in[17];
  const float* v2  = (const float*)d_in[18];
  float* out = (float*)d_out;

  const int N = in_sizes[0] / IN_FEATS;
  const int E = in_sizes[1] / 2;

  // workspace layout (floats)
  float* ws    = (float*)d_ws;
  float* bufA  = ws;                                   // N*128 (agg1, reused as agg2)
  float* h1    = ws + (long long)N * IN_FEATS;         // N*64
  float* h2    = h1 + (long long)N * HIDDEN;           // N*64
  float* sums  = h2 + (long long)N * HIDDEN;           // 512*64
  float* cnts  = sums + NGRAPH * HIDDEN;               // 512
  (void)ws_size; (void)n_in; (void)out_size;

  const int tiles  = (N + 15) / 16;
  const int mlpBlk = (tiles + 7) / 8;

  // ---- layer 1 ----
  hipMemcpyAsync(bufA, x, (size_t)N * IN_FEATS * sizeof(float),
                 hipMemcpyDeviceToDevice, stream);  // agg := x  (self term)
  {
    long long tot = (long long)E * (IN_FEATS / 4);
    int blocks = (int)((tot + 255) / 256);
    scatter_add_kernel<<<blocks, 256, 0, stream>>>(x, ei, bufA, E, IN_FEATS);
  }
  gin_mlp_kernel<IN_FEATS><<<mlpBlk, 256, 0, stream>>>(
      bufA, W1a, b1a, W1b, b1b, g1, be1, m1, v1, h1, N);

  // ---- layer 2 ----
  hipMemcpyAsync(bufA, h1, (size_t)N * HIDDEN * sizeof(float),
                 hipMemcpyDeviceToDevice, stream);  // agg := h1
  {
    long long tot = (long long)E * (HIDDEN / 4);
    int blocks = (int)((tot + 255) / 256);
    scatter_add_kernel<<<blocks, 256, 0, stream>>>(h1, ei, bufA, E, HIDDEN);
  }
  gin_mlp_kernel<HIDDEN><<<mlpBlk, 256, 0, stream>>>(
      bufA, W2a, b2a, W2b, b2b, g2, be2, m2, v2, h2, N);

  // ---- global mean pool ----
  pool_zero_kernel<<<(NGRAPH * HIDDEN + 255) / 256, 256, 0, stream>>>(sums, cnts);
  {
    long long tot = (long long)N * 16;
    int blocks = (int)((tot + 255) / 256);
    pool_sum_kernel<<<blocks, 256, 0, stream>>>(h2, batch, sums, cnts, N);
  }
  pool_div_kernel<<<(NGRAPH * HIDDEN + 255) / 256, 256, 0, stream>>>(sums, cnts, out);
}